// SpatiallySparseConv_44040594653251
// MI455X (gfx1250) — compile-verified
//
#include <hip/hip_runtime.h>

// Problem constants (match reference setup_inputs()).
#define NROW 131072
#define KOFF 27
#define PPK  32768
#define CIN  64
#define COUT 64

typedef __attribute__((ext_vector_type(2))) float v2f;
typedef __attribute__((ext_vector_type(8))) float v8f;

constexpr int TILE_P   = 128;  // pairs per tile (one block iteration)
constexpr int TILES_PB = 4;    // tiles processed per block (weight reuse)
constexpr int FSTRIDE  = 68;   // padded LDS row stride (floats); 272B = 16B-aligned

// Async gather: one B128 (16B) chunk; INST_OFFSET advances BOTH the LDS and
// the global address (CDNA5 ISA), and our strides match.
#define ASYNC_LD16(LDSOFF, GPTR, OFF)                                          \
    asm volatile("global_load_async_to_lds_b128 %0, %1, off offset:" OFF       \
                 :: "v"(LDSOFF), "v"(GPTR) : "memory")

#define WAIT_ASYNC(N) asm volatile("s_wait_asynccnt " N ::: "memory")

// ---------------------------------------------------------------------------
// out[n][c] = bias[c] for all n  (full init; harness poisons d_out)
// ---------------------------------------------------------------------------
__global__ __launch_bounds__(256)
void init_bias_kernel(const float* __restrict__ bias, float* __restrict__ out)
{
    const int tid = blockIdx.x * 256 + threadIdx.x;   // one float4 per thread
    const int c4  = tid & (COUT / 4 - 1);
    const float4 b = *(const float4*)(bias + c4 * 4);
    *(float4*)(out + (size_t)tid * 4) = b;
}

// ---------------------------------------------------------------------------
// Per-offset gather -> WMMA f32 GEMM -> atomic scatter-add, with the gather
// running on the async-to-LDS engine, double-buffered across tiles.
// Block: 256 threads = 8 waves; wave w computes pairs [w*16, w*16+16).
// ---------------------------------------------------------------------------
__global__ __launch_bounds__(256)
void sparse_conv_wmma_kernel(const float* __restrict__ feat,
                             const float* __restrict__ weight,
                             const int*   __restrict__ in_map,
                             const int*   __restrict__ out_map,
                             float*       __restrict__ out)
{
    // Weights interleaved by row pair: sW[p*128 + c*2 + (row&1)], p = row/2,
    // so each lane's B operand (rows 2p,2p+1 at column c) is a contiguous float2.
    __shared__ float sW[(CIN / 2) * (2 * COUT)];          // 16 KB
    __shared__ float sF[2][TILE_P * FSTRIDE];             // 2 x ~34 KB (double buffer)
    __shared__ int   sOut[2][TILE_P];

    const int t    = threadIdx.x;
    const int lane = t & 31;
    const int wave = t >> 5;
    const int h    = lane >> 4;   // half-wave select
    const int m    = lane & 15;   // row (A/D) / column (B,D) index within 16

    constexpr int CHUNKS = PPK / (TILE_P * TILES_PB);     // 64
    const int k       = blockIdx.x / CHUNKS;              // kernel offset 0..26
    const int chunk   = blockIdx.x % CHUNKS;
    const size_t mapB = (size_t)k * PPK + (size_t)chunk * (TILE_P * TILES_PB);

    // Async-gather ownership: thread owns row (t&127), column half (t>>7).
    const int rowid = t & (TILE_P - 1);
    const int half  = t >> 7;                             // 0 or 1 (32 floats each)

    // ---- Stage weight[k] into LDS (once per block) ----
    {
        const float* wk = weight + (size_t)k * CIN * COUT;
        #pragma unroll
        for (int i = 0; i < (CIN * COUT) / 256; ++i) {    // 16 iters
            const int idx = i * 256 + t;
            const int r = idx >> 6;
            const int c = idx & 63;
            sW[(r >> 1) * (2 * COUT) + c * 2 + (r & 1)] = wk[idx];
        }
    }

    // Issue the 8 async B128 gathers for one tile into buffer `buf`.
    auto issue_gather = [&](int buf, int tile) {
        int ir = in_map[mapB + (size_t)tile * TILE_P + rowid];
        if (ir < 0) ir = 0;   // garbage row is fine: its D-row is never scattered
        const float* g = feat + (size_t)ir * CIN + half * 32;
        // Low 32 bits of a generic LDS pointer == LDS byte offset.
        const unsigned l = (unsigned)(size_t)&sF[buf][rowid * FSTRIDE + half * 32];
        ASYNC_LD16(l, g, "0");
        ASYNC_LD16(l, g, "16");
        ASYNC_LD16(l, g, "32");
        ASYNC_LD16(l, g, "48");
        ASYNC_LD16(l, g, "64");
        ASYNC_LD16(l, g, "80");
        ASYNC_LD16(l, g, "96");
        ASYNC_LD16(l, g, "112");
    };

    // ---- Prologue: prefetch tile 0 ----
    if (t < TILE_P) sOut[0][t] = out_map[mapB + t];
    issue_gather(0, 0);

    for (int tile = 0; tile < TILES_PB; ++tile) {
        const int cur = tile & 1;

        // ---- Prefetch next tile into the other buffer, then retire current ----
        if (tile + 1 < TILES_PB) {
            if (t < TILE_P)
                sOut[cur ^ 1][t] = out_map[mapB + (size_t)(tile + 1) * TILE_P + t];
            issue_gather(cur ^ 1, tile + 1);
            WAIT_ASYNC("0x8");       // in-order: current tile's 8 loads are done
        } else {
            WAIT_ASYNC("0x0");
        }
        __syncthreads();             // all waves' gathers for `cur` visible

        // ---- GEMM: 16 pairs x 64 cin x 64 cout via V_WMMA_F32_16X16X4_F32 ----
        v8f acc0 = {}, acc1 = {}, acc2 = {}, acc3 = {};
        const float* fb = &sF[cur][0];
        const int rowm = wave * 16 + m;
        #pragma unroll
        for (int s = 0; s < CIN / 4; ++s) {
            // A 16x4: lane m holds K={0,1}; lane 16+m holds K={2,3}
            v2f a = *(const v2f*)(fb + rowm * FSTRIDE + 4 * s + 2 * h);
            // B 4x16: VGPR0 = row K=2h, VGPR1 = row K=2h+1 (interleaved pairs)
            const float* wp = &sW[(2 * s + h) * (2 * COUT)];
            v2f b0 = *(const v2f*)(wp + (0 * 16 + m) * 2);
            v2f b1 = *(const v2f*)(wp + (1 * 16 + m) * 2);
            v2f b2 = *(const v2f*)(wp + (2 * 16 + m) * 2);
            v2f b3 = *(const v2f*)(wp + (3 * 16 + m) * 2);
            acc0 = __builtin_amdgcn_wmma_f32_16x16x4_f32(false, a, false, b0,
                                                         (short)0, acc0, false, false);
            acc1 = __builtin_amdgcn_wmma_f32_16x16x4_f32(false, a, false, b1,
                                                         (short)0, acc1, false, false);
            acc2 = __builtin_amdgcn_wmma_f32_16x16x4_f32(false, a, false, b2,
                                                         (short)0, acc2, false, false);
            acc3 = __builtin_amdgcn_wmma_f32_16x16x4_f32(false, a, false, b3,
                                                         (short)0, acc3, false, false);
        }

        // ---- Scatter-add: VGPR r holds D row r+8h; lane col = 16t+m.
        // Each half-wave writes 16 consecutive floats -> 64B-contiguous atomics.
        #pragma unroll
        for (int r = 0; r < 8; ++r) {
            const int orow = sOut[cur][wave * 16 + r + 8 * h];
            if (orow >= 0) {
                float* dst = out + (size_t)orow * COUT + m;
                unsafeAtomicAdd(dst + 0,  acc0[r]);
                unsafeAtomicAdd(dst + 16, acc1[r]);
                unsafeAtomicAdd(dst + 32, acc2[r]);
                unsafeAtomicAdd(dst + 48, acc3[r]);
            }
        }
        __syncthreads();   // keep fast waves from prefetching over buffer `cur`
    }
}

// ---------------------------------------------------------------------------
extern "C" void kernel_launch(void* const* d_in, const int* in_sizes, int n_in,
                              void* d_out, int out_size, void* d_ws, size_t ws_size,
                              hipStream_t stream)
{
    const float* feat   = (const float*)d_in[0];  // [N, CIN]
    const float* weight = (const float*)d_in[1];  // [K, CIN, COUT]
    const float* bias   = (const float*)d_in[2];  // [COUT]
    const int*   in_map = (const int*)d_in[3];    // [K, P]
    const int*   out_map= (const int*)d_in[4];    // [K, P]
    float*       out    = (float*)d_out;          // [N, COUT]

    // 1) out = bias (full init of poisoned buffer)
    init_bias_kernel<<<(NROW * COUT / 4) / 256, 256, 0, stream>>>(bias, out);

    // 2) async gather -> WMMA GEMM -> scatter-add
    const int blocks = KOFF * (PPK / (TILE_P * TILES_PB));  // 27 * 64 = 1728
    sparse_conv_wmma_kernel<<<blocks, 256, 0, stream>>>(feat, weight, in_map,
                                                        out_map, out);
}